// Glm4MoeRouteTokenToExperts_1657857376736
// MI455X (gfx1250) — compile-verified
//
#include <hip/hip_runtime.h>
#include <stdint.h>

// GLM4-MoE router, thread-per-token formulation.
// Each lane owns one token: all top-k selection state is in per-lane
// registers (no cross-lane shuffles). A wave's 32 token rows (32 KB,
// contiguous in global memory) are staged into LDS by ONE Tensor Data Mover
// instruction, using TDM's pad feature to skew rows by 16 B per 1 KB so
// per-lane row reads (stride 1040 B) don't bank-conflict.
// Sorted-list maintenance uses v_med3_f32: inserting v into a descending
// sorted list is a[s] = med3(a[s-1], v, a[s]).

typedef uint32_t u32x4 __attribute__((ext_vector_type(4)));
typedef uint32_t u32x8 __attribute__((ext_vector_type(8)));

constexpr int   kExperts   = 256;
constexpr int   kGroups    = 8;
constexpr int   kTopkGroup = 4;
constexpr int   kTopK      = 8;
constexpr int   kTPW       = 32;          // tokens per wave (one per lane)
constexpr int   kRowStride = 1024 + 16;   // bytes: 1 KB row + 16 B TDM pad
constexpr float kScale     = 2.5f;

__device__ __forceinline__ float sigmoid_fast(float x) {
    // v_mul + v_exp + v_add + v_rcp
    return __builtin_amdgcn_rcpf(1.0f + __expf(-x));
}

__global__ __launch_bounds__(kTPW) void glm4_moe_route_kernel(
    const float* __restrict__ logits,   // [T, 256]
    const float* __restrict__ bias,     // [256]
    int*   __restrict__ out_idx,        // [T, 8]
    float* __restrict__ out_w,          // [T, 8]
    int n_tokens)
{
    __shared__ __align__(16) unsigned char lds[kTPW * kRowStride + 1024];
    const int lane  = (int)threadIdx.x;             // 0..31
    const int tbase = (int)blockIdx.x * kTPW;
    if (tbase >= n_tokens) return;                  // block-uniform
    const int token = tbase + lane;

    float* sbias = (float*)(lds + kTPW * kRowStride);
    float* myrow = (float*)(lds + lane * kRowStride);

    // ---- TDM: one tensor_load_to_lds moves this wave's 32 rows (32 KB,
    // contiguous in global) into LDS, inserting 16 B of pad per 1 KB row.
    {
        uint64_t ga = (uint64_t)(uintptr_t)(logits + (size_t)tbase * kExperts);
        uint32_t la = (uint32_t)(uintptr_t)lds;
        uint32_t nelem = (uint32_t)(n_tokens - tbase) * (uint32_t)kExperts;
        const uint32_t tile = (uint32_t)(kTPW * kExperts);      // 8192 elems
        if (nelem > tile) nelem = tile;                         // OOB reads -> 0

        u32x4 g0;
        g0[0] = 1u;                                             // count=1
        g0[1] = la;                                             // lds_addr
        g0[2] = (uint32_t)ga;                                   // global_addr lo
        g0[3] = ((uint32_t)(ga >> 32) & 0x01FFFFFFu)            // global_addr[56:32]
              | 0x80000000u;                                    // type=2 (image)

        u32x8 g1;
        g1[0] = (2u << 16)      // data_size = 4 B
              | (1u << 20)      // pad_enable
              | (7u << 22)      // pad_interval: 256 dwords = 1024 B
              | (3u << 25);     // pad_amount: 4 dwords = 16 B
        g1[1] = (nelem & 0xFFFFu) << 16;        // tensor_dim0[15:0]
        g1[2] = (nelem >> 16) | (1u << 16);     // tensor_dim0[31:16] | tensor_dim1=1
        g1[3] = tile << 16;                     // tile_dim0 = 8192
        g1[4] = 0u;                             // tile_dim1/2 unused
        g1[5] = tile;                           // tensor_dim0_stride (unused for 1D)
        g1[6] = 0u;
        g1[7] = 0u;
        asm volatile("tensor_load_to_lds %0, %1" :: "s"(g0), "s"(g1) : "memory");
    }

    // ---- async-copy engine: 1 KB bias row into LDS (2 x 512 B).
    {
        uint32_t bl0 = (uint32_t)(uintptr_t)sbias + (uint32_t)lane * 16u;
        uint32_t bl1 = bl0 + 512u;
        uint64_t bg0 = (uint64_t)(uintptr_t)bias + (uint64_t)((uint32_t)lane * 16u);
        uint64_t bg1 = bg0 + 512u;
        asm volatile("global_load_async_to_lds_b128 %0, %1, off" :: "v"(bl0), "v"(bg0) : "memory");
        asm volatile("global_load_async_to_lds_b128 %0, %1, off" :: "v"(bl1), "v"(bg1) : "memory");
    }

    asm volatile("s_wait_asynccnt 0x0" ::: "memory");
    asm volatile("s_wait_tensorcnt 0x0" ::: "memory");

    const float kNegInf = -__builtin_inff();

    // ---- Pass 1: sc = sigmoid(x)+bias, written back in place; per-group
    // running top-2 via med3 (2 VALU/elem) -> group scores.
    float gsum[kGroups];
    #pragma unroll
    for (int g = 0; g < kGroups; ++g) {
        float a1 = kNegInf, a2 = kNegInf;
        #pragma unroll
        for (int c = 0; c < 8; ++c) {
            float4 xv = *(const float4*)(myrow + g * 32 + c * 4);
            float4 bv = *(const float4*)(sbias + g * 32 + c * 4);  // uniform bcast
            float s0 = sigmoid_fast(xv.x) + bv.x;
            float s1 = sigmoid_fast(xv.y) + bv.y;
            float s2 = sigmoid_fast(xv.z) + bv.z;
            float s3 = sigmoid_fast(xv.w) + bv.w;
            *(float4*)(myrow + g * 32 + c * 4) = make_float4(s0, s1, s2, s3);
            a2 = __builtin_amdgcn_fmed3f(a1, s0, a2); a1 = fmaxf(a1, s0);
            a2 = __builtin_amdgcn_fmed3f(a1, s1, a2); a1 = fmaxf(a1, s1);
            a2 = __builtin_amdgcn_fmed3f(a1, s2, a2); a1 = fmaxf(a1, s2);
            a2 = __builtin_amdgcn_fmed3f(a1, s3, a2); a1 = fmaxf(a1, s3);
        }
        gsum[g] = a1 + a2;
    }

    // ---- Top-4 groups: each unordered pair compared once; rank with
    // lower-index-wins-on-tie matches jax.lax.top_k's stable ordering.
    unsigned selmask = 0;
    {
        int rk[kGroups];
        #pragma unroll
        for (int g = 0; g < kGroups; ++g) rk[g] = 0;
        #pragma unroll
        for (int g = 1; g < kGroups; ++g) {
            #pragma unroll
            for (int h = 0; h < g; ++h) {
                bool c = gsum[h] >= gsum[g];   // h beats g (h has lower index)
                rk[g] += c ? 1 : 0;
                rk[h] += c ? 0 : 1;
            }
        }
        #pragma unroll
        for (int g = 0; g < kGroups; ++g)
            selmask |= (rk[g] < kTopkGroup ? 1u : 0u) << g;
    }

    // ---- Top-8 list init: reference top_k runs over the masked array where
    // non-kept groups contribute 0.0; seed with the first masked indices at 0.
    float aV[kTopK];
    int   aI[kTopK];
    {
        const int gmin = __builtin_ctz(~selmask & 0xFFu) * 32;
        #pragma unroll
        for (int j = 0; j < kTopK; ++j) { aV[j] = 0.0f; aI[j] = gmin + j; }
    }

    // ---- Pass 2: stream the 4 selected groups (128 elems). med3-based
    // sorted insert: value chain is 1 med3/slot; index chain 1 cmp + 2 sel.
    // Ascending scan + strict '>' reproduces jax's lowest-index tie-break.
    unsigned rem = selmask;
    #pragma unroll 1
    for (int j = 0; j < kTopkGroup; ++j) {
        const int g = (int)__builtin_ctz(rem);
        rem &= rem - 1u;
        const float* rp = myrow + g * 32;
        const int idxbase = g * 32;
        #pragma unroll
        for (int c = 0; c < 8; ++c) {
            float4 sv = *(const float4*)(rp + c * 4);
            float vs[4] = {sv.x, sv.y, sv.z, sv.w};
            #pragma unroll
            for (int e = 0; e < 4; ++e) {
                const float v  = vs[e];
                const int   vi = idxbase + c * 4 + e;
                bool t[kTopK];
                #pragma unroll
                for (int s = 0; s < kTopK; ++s) t[s] = v > aV[s];   // vs OLD list
                #pragma unroll
                for (int s = kTopK - 1; s >= 1; --s) {              // descending:
                    aV[s] = __builtin_amdgcn_fmed3f(aV[s - 1], v, aV[s]); // olds intact
                    aI[s] = t[s] ? (t[s - 1] ? aI[s - 1] : vi) : aI[s];
                }
                aV[0] = fmaxf(aV[0], v);
                aI[0] = t[0] ? vi : aI[0];
            }
        }
    }

    // ---- Weights: sigmoid = sc - bias[idx] (LDS gather), normalize, scale.
    float w[kTopK];
    float denom = 1e-20f;
    #pragma unroll
    for (int j = 0; j < kTopK; ++j) {
        w[j] = aV[j] - sbias[aI[j]];
        denom += w[j];
    }

    if (token < n_tokens) {
        const float inv = kScale / denom;   // one IEEE divide
        const size_t o = (size_t)token * kTopK;
        *(int4*)(out_idx + o)       = make_int4(aI[0], aI[1], aI[2], aI[3]);
        *(int4*)(out_idx + o + 4)   = make_int4(aI[4], aI[5], aI[6], aI[7]);
        *(float4*)(out_w + o)       = make_float4(w[0]*inv, w[1]*inv, w[2]*inv, w[3]*inv);
        *(float4*)(out_w + o + 4)   = make_float4(w[4]*inv, w[5]*inv, w[6]*inv, w[7]*inv);
    }
}

extern "C" void kernel_launch(void* const* d_in, const int* in_sizes, int n_in,
                              void* d_out, int out_size, void* d_ws, size_t ws_size,
                              hipStream_t stream) {
    (void)n_in; (void)out_size; (void)d_ws; (void)ws_size;
    const float* logits = (const float*)d_in[0];   // router_logits [T, 256]
    const float* bias   = (const float*)d_in[1];   // e_score_correction_bias [256]
    const int T = in_sizes[0] / kExperts;

    // d_out: [T,8] int32 indices followed by [T,8] f32 weights (flat concat).
    int*   out_idx = (int*)d_out;
    float* out_w   = (float*)d_out + (size_t)T * kTopK;

    const int blocks = (T + kTPW - 1) / kTPW;
    glm4_moe_route_kernel<<<blocks, kTPW, 0, stream>>>(logits, bias, out_idx, out_w, T);
}